// FasterRCNN_7739531067409
// MI455X (gfx1250) — compile-verified
//
#include <hip/hip_runtime.h>

// Problem constants (match the reference).
constexpr int kB  = 8;
constexpr int kS  = 64;
constexpr int kA  = 9;
constexpr int kN  = kS * kS * kA;   // 36864 boxes per batch
constexpr int kM  = 64;             // GT boxes per batch
constexpr float kImg    = 800.0f;
constexpr float kStride = kImg / (float)kS;   // 12.5

struct BoxState {
    float x1, y1, x2, y2;   // decoded, clipped box
    float area;             // box area
    float bi, bd;           // best inter / best denominator (argmax state)
    int   bidx;             // best gt index
};

__device__ __forceinline__ BoxState decode_box(const float4 t, int n,
                                               const float* __restrict__ anch_h,
                                               const float* __restrict__ anch_w) {
    const int a  = n % kA;
    const int r  = n / kA;
    const int xg = r % kS;
    const int yg = r / kS;

    const float aw = anch_w[a];
    const float ah = anch_h[a];
    const float acx = ((float)xg + 0.5f) * kStride;
    const float acy = ((float)yg + 0.5f) * kStride;

    const float pcx = acx + t.x * aw;
    const float pcy = acy + t.y * ah;
    const float pw  = aw * expf(t.z);
    const float ph  = ah * expf(t.w);

    BoxState s;
    s.x1 = fminf(fmaxf(pcx - 0.5f * pw, 0.0f), kImg);
    s.y1 = fminf(fmaxf(pcy - 0.5f * ph, 0.0f), kImg);
    s.x2 = fminf(fmaxf(pcx + 0.5f * pw, 0.0f), kImg);
    s.y2 = fminf(fmaxf(pcy + 0.5f * ph, 0.0f), kImg);
    s.area = (s.x2 - s.x1) * (s.y2 - s.y1);
    s.bi = -1.0f;   // iteration 0 always wins
    s.bd = 1.0f;
    s.bidx = 0;
    return s;
}

__device__ __forceinline__ void iou_step(BoxState& s, const float4 g, const float area_b, const int j) {
    const float lx = fmaxf(s.x1, g.x);
    const float ly = fmaxf(s.y1, g.y);
    const float rx = fminf(s.x2, g.z);
    const float ry = fminf(s.y2, g.w);
    const float w  = fmaxf(rx - lx, 0.0f);
    const float h  = fmaxf(ry - ly, 0.0f);
    const float inter = w * h;
    const float den   = s.area + area_b - inter + 1e-9f;
    // iou_j > iou_best  <=>  inter_j * den_best > inter_best * den_j  (den > 0)
    const bool better = inter * s.bd > s.bi * den;   // strict > keeps first index on ties
    s.bi   = better ? inter : s.bi;
    s.bd   = better ? den   : s.bd;
    s.bidx = better ? j     : s.bidx;
}

__device__ __forceinline__ void finish_box(const BoxState& s, const float4 fo,
                                           size_t b, int n,
                                           float* __restrict__ out_fg,
                                           float* __restrict__ out_idx,
                                           float* __restrict__ out_ref) {
    const float best_iou = s.bi / s.bd;             // one exact division at the end
    const size_t bn = b * kN + n;
    out_fg[bn]  = (best_iou > 0.5f) ? 1.0f : 0.0f;
    out_idx[bn] = (float)s.bidx;

    const float w0 = s.x2 - s.x1;
    const float h0 = s.y2 - s.y1;
    const float rx1 = fo.x * w0 + s.x1;
    const float ry1 = fo.y * h0 + s.y1;
    const float rx2 = rx1 + expf(fo.z) * w0;
    const float ry2 = ry1 + expf(fo.w) * h0;
    *(float4*)(out_ref + bn * 4) = make_float4(rx1, ry1, rx2, ry2);
}

__global__ __launch_bounds__(256) void frcnn_match_kernel(
    const float* __restrict__ rpn_off,   // (B, N, 4)
    const float* __restrict__ frc_off,   // (B, N, 4)
    const float* __restrict__ gt,        // (B, M, 4)
    const float* __restrict__ anch_h,    // (A)
    const float* __restrict__ anch_w,    // (A)
    float* __restrict__ out_pred,        // (B, N, 4)
    float* __restrict__ out_fg,          // (B, N)
    float* __restrict__ out_idx,         // (B, N)
    float* __restrict__ out_ref)         // (B, N, 4)
{
    __shared__ float4 s_gt[kM];
    __shared__ float  s_area[kM];

    const int b   = blockIdx.y;
    const int tid = threadIdx.x;
    // Two boxes per thread: amortizes the wave-uniform LDS reads over 2 IoU
    // chains and feeds the VOPD dual-issue pipes with independent work.
    const int n0 = blockIdx.x * 512 + tid;
    const int n1 = n0 + 256;

    // ---- Stage this batch's 64 GT boxes into LDS with the CDNA5 async path ----
    // tid<64 == waves 0 and 1 fully active (wave32), so EXEC is all-ones there.
    if (tid < kM) {
        const float* gbase = gt + (size_t)b * kM * 4;
        // Low 32 bits of a generic pointer into the LDS aperture are the LDS byte offset.
        unsigned lds_off = (unsigned)(unsigned long long)(uintptr_t)(&s_gt[tid]);
        unsigned goff    = (unsigned)tid * 16u;
        asm volatile("global_load_async_to_lds_b128 %0, %1, %2 offset:0"
                     :
                     : "v"(lds_off), "v"(goff), "s"(gbase)
                     : "memory");
    }
    asm volatile("s_wait_asynccnt 0x0" ::: "memory");
    __syncthreads();

    // Precompute GT areas once per block.
    if (tid < kM) {
        float4 g = s_gt[tid];
        s_area[tid] = (g.z - g.x) * (g.w - g.y);
    }
    __syncthreads();

    // ---- Decode the two RPN boxes ----
    const size_t base0 = ((size_t)b * kN + n0) * 4;
    const size_t base1 = ((size_t)b * kN + n1) * 4;
    const float4 t0  = *(const float4*)(rpn_off + base0);
    const float4 t1  = *(const float4*)(rpn_off + base1);
    const float4 fo0 = *(const float4*)(frc_off + base0);
    const float4 fo1 = *(const float4*)(frc_off + base1);

    BoxState s0 = decode_box(t0, n0, anch_h, anch_w);
    BoxState s1 = decode_box(t1, n1, anch_h, anch_w);

    *(float4*)(out_pred + base0) = make_float4(s0.x1, s0.y1, s0.x2, s0.y2);
    *(float4*)(out_pred + base1) = make_float4(s1.x1, s1.y1, s1.x2, s1.y2);

    // ---- IoU max/argmax over 64 GT boxes; LDS reads shared by both boxes ----
#pragma unroll 4
    for (int j = 0; j < kM; ++j) {
        const float4 g  = s_gt[j];
        const float ab  = s_area[j];
        iou_step(s0, g, ab, j);
        iou_step(s1, g, ab, j);
    }

    finish_box(s0, fo0, (size_t)b, n0, out_fg, out_idx, out_ref);
    finish_box(s1, fo1, (size_t)b, n1, out_fg, out_idx, out_ref);
}

extern "C" void kernel_launch(void* const* d_in, const int* in_sizes, int n_in,
                              void* d_out, int out_size, void* d_ws, size_t ws_size,
                              hipStream_t stream) {
    (void)in_sizes; (void)n_in; (void)out_size; (void)d_ws; (void)ws_size;
    const float* rpn = (const float*)d_in[0];   // (B,S,S,A,4)
    const float* frc = (const float*)d_in[1];   // (B,N,4)
    const float* gt  = (const float*)d_in[2];   // (B,M,4)
    const float* ah  = (const float*)d_in[3];   // (A)
    const float* aw  = (const float*)d_in[4];   // (A)

    float* out = (float*)d_out;
    const size_t BN = (size_t)kB * kN;
    float* out_pred = out;            // BN*4
    float* out_fg   = out + BN * 4;   // BN
    float* out_idx  = out + BN * 5;   // BN
    float* out_ref  = out + BN * 6;   // BN*4

    dim3 grid(kN / 512, kB);          // 72 x 8 blocks, 256 threads (8 waves), 2 boxes/thread
    frcnn_match_kernel<<<grid, 256, 0, stream>>>(rpn, frc, gt, ah, aw,
                                                 out_pred, out_fg, out_idx, out_ref);
}